// OptimizedPhotonicLayer_21242908246430
// MI455X (gfx1250) — compile-verified
//
#include <hip/hip_runtime.h>

// ---------------------------------------------------------------------------
// Photonic mesh layer for MI455X (gfx1250, wave32).
//
// Phase A: build U (256x256 complex) from 32640 Givens rotations.
//   - one thread per column of U (columns evolve independently)
//   - per-layer recurrence carried in registers: u[p] = c*v + i*s*a[p+1],
//     v' = c*a[p+1] + i*s*v  (a[] = pre-layer values, read-only)
//   - column lives in LDS [row][col32] (float2) -> 64-bank conflict-free
// Phase B: out = in @ U^T as two real fp32 GEMMs via V_WMMA_F32_16X16X4_F32.
//   - 4.3 GFLOP, 50MB HBM traffic -> memory-bound (~2us @ 23.3 TB/s)
//   - block = 8 waves sharing one ntile: 32KB B tile staged to LDS once via
//     GLOBAL_LOAD_ASYNC_TO_LDS_B64 (ASYNCcnt), 8x less L2 traffic
//   - B fragment reads are ds_load_b64, conflict-free across all 64 banks
// ---------------------------------------------------------------------------

typedef __attribute__((ext_vector_type(2))) float v2f;
typedef __attribute__((ext_vector_type(8))) float v8f;

typedef int pm_i64v __attribute__((vector_size(8)));                 // 64-bit payload
typedef __attribute__((address_space(1))) pm_i64v* pm_gp64;          // global ptr
typedef __attribute__((address_space(3))) pm_i64v* pm_lp64;          // LDS ptr

#define PM_SIZE   256
#define PM_NPH    32640      // 256*255/2
#define PM_BATCH  16384
#define PM_COLS   32         // columns per build block (64KB LDS)

#if defined(__AMDGCN__) && __has_builtin(__builtin_amdgcn_global_load_async_to_lds_b64)
#define PM_ASYNC 1
#else
#define PM_ASYNC 0
#endif

// ---------------------------------------------------------------- cs table
__global__ __launch_bounds__(256) void pm_cs_kernel(
    const float* __restrict__ phases, float2* __restrict__ cs) {
  int k = blockIdx.x * 256 + threadIdx.x;
  if (k < PM_NPH) {
    float h = 0.5f * phases[k];
    cs[k] = make_float2(cosf(h), sinf(h));
  }
}

// ------------------------------------------------------------- build U^T
// Writes Bre/Bim in K-pair-packed layout:
//   elem(k=j, n=i) at float offset (j>>1)*512 + i*2 + (j&1)
__global__ __launch_bounds__(PM_COLS) void pm_build_kernel(
    const float2* __restrict__ cs,
    float* __restrict__ Bre, float* __restrict__ Bim) {
  __shared__ float2 col[PM_SIZE][PM_COLS];   // 64 KB, bank-conflict free
  const int t = threadIdx.x;                 // 0..31 (one wave)
  const int j = blockIdx.x * PM_COLS + t;    // my column of U

  for (int p = 0; p < PM_SIZE; ++p)
    col[p][t] = make_float2(p == j ? 1.0f : 0.0f, 0.0f);

  int k = 0;
  for (int l = 0; l < PM_SIZE - 1; ++l) {
    const int m = PM_SIZE - 1 - l;           // rotations in this layer
    float2 v = col[0][t];                    // register carry
    #pragma unroll 4
    for (int p = 0; p < m; ++p) {
      const float2 csk = cs[k + p];          // uniform across lanes
      const float2 a   = col[p + 1][t];      // pre-layer value
      float2 u, vn;
      u.x  = fmaf(csk.x, v.x, -csk.y * a.y); // (c*v + i*s*a).re
      u.y  = fmaf(csk.x, v.y,  csk.y * a.x); // (c*v + i*s*a).im
      vn.x = fmaf(csk.x, a.x, -csk.y * v.y); // (c*a + i*s*v).re
      vn.y = fmaf(csk.x, a.y,  csk.y * v.x); // (c*a + i*s*v).im
      col[p][t] = u;
      v = vn;
    }
    col[m][t] = v;
    k += m;
  }

  // B[k=j][n=i] = U[i][j], packed in K pairs for b64 fragment loads
  for (int i = 0; i < PM_SIZE; ++i) {
    const float2 val = col[i][t];
    const int off = (j >> 1) * (2 * PM_SIZE) + i * 2 + (j & 1);
    Bre[off] = val.x;
    Bim[off] = val.y;
  }
}

// ------------------------------------------------------------- WMMA GEMM
// Block = 8 waves, one ntile per block, 8 consecutive mtiles.
// B tile (2 planes x 128 kpairs x 16 cols, float2 = 32KB) staged in LDS.
__global__ __launch_bounds__(256) void pm_gemm_kernel(
    const float* __restrict__ in,
    const float* __restrict__ BreP, const float* __restrict__ BimP,
    float2* __restrict__ out) {
  __shared__ float2 ldsB[2][PM_SIZE / 2][16];   // 32 KB

  const int tid   = threadIdx.x;
  const int ntile = blockIdx.x & 15;            // 0..15   (256/16)
  const int mgrp  = blockIdx.x >> 4;            // 0..127  (1024/8)
  const int wave  = tid >> 5;
  const int lane  = tid & 31;
  const int mtile = mgrp * 8 + wave;            // 0..1023 (batch/16)
  const int hl    = lane >> 4;                  // half-wave: K pair lo/hi
  const int lidx  = lane & 15;

  // ---- stage B tile (both planes) into LDS: 4096 float2, 16 per thread ----
  const float2* __restrict__ bre2 = (const float2*)BreP;
  const float2* __restrict__ bim2 = (const float2*)BimP;
  #pragma unroll
  for (int i = 0; i < 16; ++i) {
    const int e     = tid + 256 * i;            // 0..4095
    const int plane = e >> 11;
    const int rem   = e & 2047;
    const int kp    = rem >> 4;
    const int c     = rem & 15;
    const float2* src =
        (plane ? bim2 : bre2) + (kp * PM_SIZE + ntile * 16 + c);
#if PM_ASYNC
    __builtin_amdgcn_global_load_async_to_lds_b64(
        (pm_gp64)(uintptr_t)src,
        (pm_lp64)(pm_i64v*)&ldsB[plane][kp][c],
        0, 0);
#else
    ldsB[plane][kp][c] = *src;
#endif
  }
#if PM_ASYNC
  asm volatile("s_wait_asynccnt 0x0" ::: "memory");
#endif
  __syncthreads();

  const float2* __restrict__ aRow =
      (const float2*)(in + (mtile * 16 + lidx) * PM_SIZE);
  const int ncol = ntile * 16 + lidx;

  v8f accRe = {};
  v8f accIm = {};

  #pragma unroll 4
  for (int kb = 0; kb < PM_SIZE / 4; ++kb) {
    const int kp = kb * 2 + hl;                 // float2 index of K pair
    const float2 af  = aRow[kp];                // A: (M=lidx, K=2kp,2kp+1)
    const float2 brf = ldsB[0][kp][lidx];       // ds_load_b64
    const float2 bif = ldsB[1][kp][lidx];       // ds_load_b64
    v2f a;  a.x = af.x;  a.y = af.y;
    v2f br; br.x = brf.x; br.y = brf.y;
    v2f bi; bi.x = bif.x; bi.y = bif.y;
    accRe = __builtin_amdgcn_wmma_f32_16x16x4_f32(
        false, a, false, br, (short)0, accRe, false, false);
    accIm = __builtin_amdgcn_wmma_f32_16x16x4_f32(
        false, a, false, bi, (short)0, accIm, false, false);
  }

  // D layout: VGPR r -> (M = r + 8*hl, N = lane%16)
  #pragma unroll
  for (int r = 0; r < 8; ++r) {
    const int brow = mtile * 16 + hl * 8 + r;
    out[brow * PM_SIZE + ncol] = make_float2(accRe[r], accIm[r]);
  }
}

// ---------------------------------------------------------------- launch
extern "C" void kernel_launch(void* const* d_in, const int* in_sizes, int n_in,
                              void* d_out, int out_size, void* d_ws, size_t ws_size,
                              hipStream_t stream) {
  const float* inputs = (const float*)d_in[0];   // [16384,256] f32
  const float* phases = (const float*)d_in[1];   // [32640] f32

  // workspace: cs table (256KB-pad) | Bre packed (256KB) | Bim packed (256KB)
  float*  ws  = (float*)d_ws;
  float2* cs  = (float2*)ws;          // 32640 float2 = 261120 B
  float*  Bre = ws + 65536;           // byte offset 262144
  float*  Bim = ws + 131072;          // byte offset 524288

  pm_cs_kernel<<<(PM_NPH + 255) / 256, 256, 0, stream>>>(phases, cs);
  pm_build_kernel<<<PM_SIZE / PM_COLS, PM_COLS, 0, stream>>>(cs, Bre, Bim);

  const int blocks = (PM_BATCH / 16 / 8) * (PM_SIZE / 16);   // 2048
  pm_gemm_kernel<<<blocks, 256, 0, stream>>>(inputs, Bre, Bim,
                                             (float2*)d_out);
}